// SpatialCoherenceConv_11312943857916
// MI455X (gfx1250) — compile-verified
//
#include <hip/hip_runtime.h>
#include <hip/hip_bf16.h>

typedef float v2f __attribute__((ext_vector_type(2)));
typedef float v8f __attribute__((ext_vector_type(8)));
typedef int   v4i __attribute__((vector_size(4 * sizeof(int))));

// Problem constants (match reference)
#define BATCH   8
#define CHANS   32
#define IMH     512
#define IMW     512
#define HALO    4

// Tiling
#define TILE_H  32
#define TILE_W  64
#define XT_H    (TILE_H + 2*HALO)   // 40
#define XT_W    (TILE_W + 2*HALO)   // 72 floats (288 B rows, 16B-aligned)
#define NTHREADS 256
#define NWAVES   (NTHREADS/32)

// ---- gfx1250 async global->LDS path (guarded; falls back to float4 copies) ----
#define GASV(p) ((__attribute__((address_space(1))) v4i*)(p))
#define LASV(p) ((__attribute__((address_space(3))) v4i*)(p))

#if defined(__has_builtin)
#if __has_builtin(__builtin_amdgcn_global_load_async_to_lds_b128)
#define HAVE_ASYNC_LDS 1
#define ASYNC_LDS_B128(g, l) __builtin_amdgcn_global_load_async_to_lds_b128(GASV(g), LASV(l), 0, 0)
#endif
#if __has_builtin(__builtin_amdgcn_s_wait_asynccnt)
#define WAIT_ASYNC() __builtin_amdgcn_s_wait_asynccnt(0)
#else
#define WAIT_ASYNC() asm volatile("s_wait_asynccnt 0" ::: "memory")
#endif
#endif

#ifndef WAIT_ASYNC
#define WAIT_ASYNC() do {} while (0)
#endif

__global__ void scc_init(double* acc) {
    if (threadIdx.x == 0 && blockIdx.x == 0) acc[0] = 0.0;
}

__global__ __launch_bounds__(NTHREADS)
void scc_main(const float* __restrict__ img, double* __restrict__ acc) {
    __shared__ __align__(16) float XT[XT_H * XT_W];   // input tile with halo
    __shared__ float RS[XT_H * TILE_W];               // 9-tap row sums of x
    __shared__ float QS[XT_H * TILE_W];               // 9-tap row sums of x^2
    __shared__ float wsum[NWAVES];

    const int tid   = threadIdx.x;
    const int tileX = blockIdx.x;          // 0..7   (W / 64)
    const int tileY = blockIdx.y;          // 0..15  (H / 32)
    const int plane = blockIdx.z;          // 0..255 (B*C)

    const long long planeOff = (long long)plane * (IMH * IMW);
    const int gy0 = tileY * TILE_H - HALO;
    const int gx0 = tileX * TILE_W - HALO;

    // ---- Phase 1: load halo tile into LDS ----
    const bool interior = (gy0 >= 0) && (gy0 + XT_H <= IMH) &&
                          (gx0 >= 0) && (gx0 + XT_W <= IMW);
    if (interior) {
        // 72 floats/row = 18 x b128; row start is 16B aligned.
        const int NV4 = XT_H * (XT_W / 4);   // 720
        for (int idx = tid; idx < NV4; idx += NTHREADS) {
            const int r  = idx / (XT_W / 4);
            const int c4 = idx - r * (XT_W / 4);
            const float* gsrc = img + planeOff + (long long)(gy0 + r) * IMW + gx0 + c4 * 4;
            float* ldst = &XT[r * XT_W + c4 * 4];
#ifdef HAVE_ASYNC_LDS
            ASYNC_LDS_B128(gsrc, ldst);      // global_load_async_to_lds_b128
#else
            *(float4*)ldst = *(const float4*)gsrc;
#endif
        }
#ifdef HAVE_ASYNC_LDS
        WAIT_ASYNC();                        // s_wait_asynccnt 0
#endif
    } else {
        // Boundary tile: scalar loads with zero padding.
        for (int idx = tid; idx < XT_H * XT_W; idx += NTHREADS) {
            const int r = idx / XT_W;
            const int c = idx - r * XT_W;
            const int gh = gy0 + r;
            const int gw = gx0 + c;
            float v = 0.0f;
            if ((unsigned)gh < (unsigned)IMH && (unsigned)gw < (unsigned)IMW) {
                v = img[planeOff + (long long)gh * IMW + gw];
            }
            XT[idx] = v;
        }
    }
    __syncthreads();

    // ---- Phase 2: horizontal 9-tap sums of x and x^2 ----
    for (int idx = tid; idx < XT_H * TILE_W; idx += NTHREADS) {
        const int r = idx >> 6;            // / TILE_W
        const int j = idx & (TILE_W - 1);
        const int base = r * XT_W + j;
        float s = 0.0f, q = 0.0f;
#pragma unroll
        for (int t = 0; t < 9; ++t) {
            const float xv = XT[base + t];
            s += xv;
            q = fmaf(xv, xv, q);
        }
        RS[idx] = s;
        QS[idx] = q;
    }
    __syncthreads();

    // ---- Phase 3: vertical 9-tap sums + fused pointwise expression ----
    // v = 81*x^2 - 2*x*S(x) + S(x^2)
    float part = 0.0f;
    for (int idx = tid; idx < TILE_H * TILE_W; idx += NTHREADS) {
        const int i = idx >> 6;
        const int j = idx & (TILE_W - 1);
        float S = 0.0f, Sq = 0.0f;
        const int rbase = i * TILE_W + j;
#pragma unroll
        for (int t = 0; t < 9; ++t) {
            S  += RS[rbase + t * TILE_W];
            Sq += QS[rbase + t * TILE_W];
        }
        const float x = XT[(i + HALO) * XT_W + (j + HALO)];
        part += fmaf(81.0f * x, x, fmaf(-2.0f * x, S, Sq));
    }

    // ---- Phase 4: wave reduction via V_WMMA_F32_16X16X4_F32 ----
    // A(16x4): lane l<16 holds A[l,0..1], lane l>=16 holds A[l-16,2..3].
    // With a = {part, 0} every lane partial appears exactly once in A.
    // B = all ones (layout-irrelevant) => D[m,n] = p_m + p_{m+16} for all n.
    // C/D layout: lane n holds rows 0..7 of column n, lane n+16 rows 8..15
    // -> sum own 8 regs, add xor-16 partner = wave total.
    v2f a; a.x = part; a.y = 0.0f;
    v2f b1; b1.x = 1.0f; b1.y = 1.0f;
    v8f cz = {};
    v8f d = __builtin_amdgcn_wmma_f32_16x16x4_f32(
        /*neg_a=*/false, a, /*neg_b=*/false, b1,
        /*c_mod=*/(short)0, cz, /*reuse_a=*/false, /*reuse_b=*/false);
    float q = d[0] + d[1] + d[2] + d[3] + d[4] + d[5] + d[6] + d[7];
    q += __shfl_xor(q, 16, 32);   // full wave32 sum in every lane

    const int lane = tid & 31;
    const int wave = tid >> 5;
    if (lane == 0) wsum[wave] = q;
    __syncthreads();

    // ---- Phase 5: one fp64 atomic per block ----
    if (tid == 0) {
        float t = 0.0f;
#pragma unroll
        for (int i = 0; i < NWAVES; ++i) t += wsum[i];
        atomicAdd(acc, (double)t);
    }
}

__global__ void scc_finalize(const double* acc, float* out) {
    if (threadIdx.x == 0 && blockIdx.x == 0) {
        out[0] = (float)(acc[0] * (1.0 / (double)BATCH));
    }
}

extern "C" void kernel_launch(void* const* d_in, const int* in_sizes, int n_in,
                              void* d_out, int out_size, void* d_ws, size_t ws_size,
                              hipStream_t stream) {
    (void)in_sizes; (void)n_in; (void)out_size; (void)ws_size;
    const float* img = (const float*)d_in[0];
    double* acc = (double*)d_ws;
    float* out = (float*)d_out;

    scc_init<<<1, 32, 0, stream>>>(acc);

    dim3 grid(IMW / TILE_W, IMH / TILE_H, BATCH * CHANS);  // 8 x 16 x 256
    scc_main<<<grid, NTHREADS, 0, stream>>>(img, acc);

    scc_finalize<<<1, 32, 0, stream>>>(acc, out);
}